// KNN_60610578481805
// MI455X (gfx1250) — compile-verified
//
#include <hip/hip_runtime.h>

typedef float v2f __attribute__((ext_vector_type(2)));
typedef float v8f __attribute__((ext_vector_type(8)));

#define B_       4
#define M_       8192
#define N_       8192
#define C_       64
#define KNB      16
#define M_TILE   128
#define N_CHUNK  64
#define THREADS  128
#define LSTRIDE  68          // pad: 16B-aligned b128, bank stride 4 for b64 frags
#define N_CHUNKS (N_ / N_CHUNK)

// ---- gfx1250 async global->LDS copy (ASYNCcnt-tracked DMA) ----
__device__ __forceinline__ void async_b128(const void* gbase, unsigned voff_bytes,
                                           void* lds_ptr) {
  // generic 'shared' pointers are {SHARED_BASE_hi32 | lds_offset_lo32}; the
  // low 32 bits are the wave-relative LDS byte address the instruction wants.
  unsigned lds = (unsigned)(unsigned long long)lds_ptr;
  asm volatile("global_load_async_to_lds_b128 %0, %1, %2 offset:0"
               :: "v"(lds), "v"(voff_bytes), "s"(gbase)
               : "memory");
}

template <int N>
__device__ __forceinline__ void wait_asynccnt() {
  asm volatile("s_wait_asynccnt %0" :: "i"(N) : "memory");
}

__launch_bounds__(THREADS, 1)
__global__ void knn_wmma_kernel(const float* __restrict__ query,
                                const float* __restrict__ support,
                                float* __restrict__ out_vals,
                                int* __restrict__ out_idx) {
  extern __shared__ float smem[];
  float* sQ  = smem;                          // M_TILE  * LSTRIDE
  float* sS0 = sQ  + M_TILE  * LSTRIDE;       // N_CHUNK * LSTRIDE (buf 0)
  float* sS1 = sS0 + N_CHUNK * LSTRIDE;       // N_CHUNK * LSTRIDE (buf 1)
  float* sD  = sS1 + N_CHUNK * LSTRIDE;       // M_TILE  * LSTRIDE
  float* sqQ = sD  + M_TILE  * LSTRIDE;       // M_TILE
  float* sqS = sqQ + M_TILE;                  // N_CHUNK

  const int t    = threadIdx.x;
  const int lane = t & 31;
  const int wave = t >> 5;
  const int b    = blockIdx.y;
  const int mt   = blockIdx.x;

  const float* gQ = query   + ((size_t)b * M_ + (size_t)mt * M_TILE) * C_;
  const float* gS = support + (size_t)b * N_ * C_;

  // ---- async-stage Q tile (32KB, 16 instr/wave) ----
#pragma unroll
  for (int p = 0; p < 16; ++p) {
    int i = t + p * THREADS;                   // float4 index
    int r = i >> 4;                            // row (C_/4 == 16 float4 per row)
    int c = (i & 15) << 2;                     // col
    async_b128(gQ, (unsigned)(i * 16), &sQ[r * LSTRIDE + c]);
  }
  // ---- async-stage S chunk 0 into buf0 (16KB, 8 instr/wave) ----
#pragma unroll
  for (int p = 0; p < 8; ++p) {
    int i = t + p * THREADS;
    int r = i >> 4;
    int c = (i & 15) << 2;
    async_b128(gS, (unsigned)(i * 16), &sS0[r * LSTRIDE + c]);
  }
  wait_asynccnt<8>();     // Q's 16 (oldest) complete; chunk0's 8 may remain
  __syncthreads();

  // ---- per-row |q|^2 ----
  {
    const float* qr = &sQ[t * LSTRIDE];
    float s = 0.f;
#pragma unroll
    for (int c = 0; c < C_; ++c) s = fmaf(qr[c], qr[c], s);
    sqQ[t] = s;
  }
  __syncthreads();        // publish sqQ so every lane can cache its slab values

  // ---- WMMA fragment addressing (V_WMMA_F32_16X16X4_F32 layout) ----
  const int lrow  = lane & 15;
  const int khalf = (lane >> 4) << 1;
  const int mhalf = (lane >> 4) << 3;

  // cache this lane's |q|^2 epilogue operands for both M-slabs in registers
  // (sqQ is immutable from here on) -> no LDS round-trips in the epilogue
  float qv[2][8];
#pragma unroll
  for (int half = 0; half < 2; ++half) {
    const int mb = (((wave + 4 * half) << 4) + mhalf);
    float4 q0 = *(const float4*)&sqQ[mb];
    float4 q1 = *(const float4*)&sqQ[mb + 4];
    qv[half][0] = q0.x; qv[half][1] = q0.y; qv[half][2] = q0.z; qv[half][3] = q0.w;
    qv[half][4] = q1.x; qv[half][5] = q1.y; qv[half][6] = q1.z; qv[half][7] = q1.w;
  }

  // ---- register-resident sorted top-16 (ascending on d^2) ----
  float lv[KNB];
  int   li[KNB];
#pragma unroll
  for (int j = 0; j < KNB; ++j) { lv[j] = 3.402823466e38f; li[j] = 0; }
  float worst = lv[KNB - 1];

  for (int chunk = 0; chunk < N_CHUNKS; ++chunk) {
    float* cur = (chunk & 1) ? sS1 : sS0;

    // ---- issue async for next chunk into the other buffer, then wait for cur.
    // The other buffer was last read by WMMA(chunk-1), which completed before
    // the barrier ahead of selection(chunk-1) that every wave has passed.
    if (chunk + 1 < N_CHUNKS) {
      float* nxt = (chunk & 1) ? sS0 : sS1;
      const float* gSc = gS + (size_t)(chunk + 1) * N_CHUNK * C_;
#pragma unroll
      for (int p = 0; p < 8; ++p) {
        int i = t + p * THREADS;
        int r = i >> 4;
        int c = (i & 15) << 2;
        async_b128(gSc, (unsigned)(i * 16), &nxt[r * LSTRIDE + c]);
      }
      wait_asynccnt<8>();   // in-order: cur's 8 done, next's 8 still in flight
    } else {
      wait_asynccnt<0>();
    }
    __syncthreads();        // cur visible to all waves

    // ---- per-column |s|^2 ----
    if (t < N_CHUNK) {
      const float* sr = &cur[t * LSTRIDE];
      float s = 0.f;
#pragma unroll
      for (int c = 0; c < C_; ++c) s = fmaf(sr[c], sr[c], s);
      sqS[t] = s;
    }
    __syncthreads();

    // ---- WMMA: each wave owns 2 M-slabs ----
#pragma unroll
    for (int half = 0; half < 2; ++half) {
      const int m0 = (wave + 4 * half) << 4;           // 0..112
      const int mb = m0 + mhalf;
      const float* qrow = &sQ[(m0 + lrow) * LSTRIDE];
      v2f a[16];
#pragma unroll
      for (int kk = 0; kk < 16; ++kk)
        a[kk] = *(const v2f*)(qrow + (kk << 2) + khalf);

#pragma unroll
      for (int ni = 0; ni < N_CHUNK / 16; ++ni) {
        const int n0 = ni << 4;
        const float* srow = &cur[(n0 + lrow) * LSTRIDE];
        v8f acc = {0.f, 0.f, 0.f, 0.f, 0.f, 0.f, 0.f, 0.f};
#pragma unroll
        for (int kk = 0; kk < 16; ++kk) {              // K = 64 in steps of 4
          v2f bv = *(const v2f*)(srow + (kk << 2) + khalf);
          acc = __builtin_amdgcn_wmma_f32_16x16x4_f32(
              false, a[kk], false, bv, (short)0, acc, false, false);
        }
        // epilogue: d2 = |q|^2 + |s|^2 - 2 q.s ; all operands in registers,
        // 8 independent b32 stores, no LDS round-trips
        const float sqs = sqS[n0 + lrow];
        const int   nn  = n0 + lrow;
#pragma unroll
        for (int r = 0; r < 8; ++r) {
          float d2 = fmaf(-2.f, acc[r], qv[half][r] + sqs);
          sD[(mb + r) * LSTRIDE + nn] = d2;
        }
      }
    }
    __syncthreads();

    // ---- selection: thread t owns query row t; float4 reads + min4 early-out ----
    const float4* drow  = (const float4*)&sD[t * LSTRIDE];
    const int     nbase = chunk * N_CHUNK;
    for (int n4 = 0; n4 < N_CHUNK / 4; ++n4) {
      float4 d4 = drow[n4];
      float  m01 = fminf(d4.x, d4.y);
      float  m23 = fminf(d4.z, d4.w);
      if (fminf(m01, m23) < worst) {                   // rare path
        float dd[4] = {d4.x, d4.y, d4.z, d4.w};
#pragma unroll
        for (int e = 0; e < 4; ++e) {
          float d = dd[e];
          if (d < worst) {
            float v  = d;
            int   id = nbase + n4 * 4 + e;
#pragma unroll
            for (int j = 0; j < KNB; ++j) {
              if (v < lv[j]) {
                float tv = lv[j]; lv[j] = v;  v  = tv;
                int   ti = li[j]; li[j] = id; id = ti;
              }
            }
            worst = lv[KNB - 1];
          }
        }
      }
    }
  }

  // ---- finalize: sqrt only the winning 16 per row ----
  const size_t gm   = (size_t)mt * M_TILE + t;
  const size_t base = ((size_t)b * M_ + gm) * KNB;
#pragma unroll
  for (int j = 0; j < KNB; ++j) {
    float v = lv[j] > 0.f ? lv[j] : 0.f;
    out_vals[base + j] = __builtin_sqrtf(v);
    out_idx[base + j]  = li[j];
  }
}

extern "C" void kernel_launch(void* const* d_in, const int* in_sizes, int n_in,
                              void* d_out, int out_size, void* d_ws, size_t ws_size,
                              hipStream_t stream) {
  const float* query   = (const float*)d_in[0];
  const float* support = (const float*)d_in[1];
  float* vals = (float*)d_out;
  int*   idx  = (int*)(vals + (size_t)B_ * M_ * KNB);  // outputs concatenated: vals then idx

  dim3 grid(M_ / M_TILE, B_);
  size_t shmem = (size_t)((M_TILE + 2 * N_CHUNK + M_TILE) * LSTRIDE +
                          M_TILE + N_CHUNK) * sizeof(float);
  knn_wmma_kernel<<<grid, THREADS, shmem, stream>>>(query, support, vals, idx);
}